// HolographicToposFunctor_18794776887933
// MI455X (gfx1250) — compile-verified
//
#include <hip/hip_runtime.h>
#include <hip/hip_bf16.h>

#define RES 96
#define CENTER 48
#define NPIX (RES * RES)        // 9216
#define TILES (NPIX / 16)       // 576 source tiles of 16 pixels
#define SPLITS 8                // source tiles per wave = TILES/SPLITS = 72
#define TPW (TILES / SPLITS)    // 72
#define EPSF 1e-5f

typedef float v2f __attribute__((ext_vector_type(2)));
typedef float v8f __attribute__((ext_vector_type(8)));

// One block (8 wave32) per 16-target tile. Each wave covers 72 of the 576
// source tiles; partial row-sums are combined through LDS at the end.
// d2 for a 16x16 (target x source) tile comes from one V_WMMA_F32_16X16X4_F32:
//   A[m][0..3] = ( i_t, j_t, i_t^2+j_t^2, 1 )
//   B[0..3][n] = ( -2 i_s, -2 j_s, 1, i_s^2+j_s^2 )
//   D[m][n]    = (i_t-i_s)^2 + (j_t-j_s)^2   (exact: all integers < 2^24)
__global__ __launch_bounds__(256) void holo_wmma_kernel(
    const float* __restrict__ boundary, float* __restrict__ out) {
  __shared__ float wlds[NPIX];           // boundary staged via async DMA
  __shared__ float redbuf[SPLITS * 16];  // per-wave partial row sums

  // ---- stage boundary into LDS with the gfx1250 async global->LDS path ----
  // dsaddr = low 32 bits of the generic shared pointer (flat aperture rule),
  // global addr = SADDR(base) + VADDR(byte offset).
  {
    const unsigned long long gbase = (unsigned long long)(uintptr_t)boundary;
    for (int idx = threadIdx.x; idx < NPIX; idx += 256) {
      unsigned int ldsoff = (unsigned int)(uintptr_t)(&wlds[idx]);
      unsigned int goff   = (unsigned int)idx * 4u;
      asm volatile("global_load_async_to_lds_b32 %0, %1, %2"
                   :: "v"(ldsoff), "v"(goff), "s"(gbase)
                   : "memory");
    }
    asm volatile("s_wait_asynccnt 0" ::: "memory");
  }
  __syncthreads();

  const int lane  = threadIdx.x & 31;
  const int wave  = threadIdx.x >> 5;   // source split 0..7
  const int tileT = blockIdx.x;         // target tile 0..575
  const int n  = lane & 15;             // A row m / B,D column n
  const int hf = lane >> 4;             // K half: 0 -> K=0,1 ; 1 -> K=2,3

  // ---- A operand: target row m = n of this tile ----
  const int tA = tileT * 16 + n;
  const int ti = tA / RES;
  const int tj = tA - ti * RES;
  v2f a;
  if (hf == 0) { a.x = (float)ti;                  a.y = (float)tj; }
  else         { a.x = (float)(ti * ti + tj * tj); a.y = 1.0f;      }

  // ---- per-lane z^2 for the 8 D-rows this lane holds: m = k + 8*hf ----
  float z2[8];
  #pragma unroll
  for (int k = 0; k < 8; ++k) {
    int m = k + 8 * hf;
    int t = tileT * 16 + m;
    int i = t / RES, j = t - i * RES;
    float di = (float)(i - CENTER), dj = (float)(j - CENTER);
    float r  = __builtin_amdgcn_sqrtf(di * di + dj * dj);
    float z  = (float)CENTER - r + EPSF;
    z2[k] = z * z;
  }

  float acc[8];
  #pragma unroll
  for (int k = 0; k < 8; ++k) acc[k] = 0.0f;

  // This wave's source range: s = wave*1152 + it*16 + n. 1152 = 12*96, so the
  // starting coordinates are (bi, bj) = (wave*12, n); step of 16 divides 96.
  int bi = wave * 12;
  int bj = n;
  int sIdx = wave * TPW * 16 + n;
  for (int it = 0; it < TPW; ++it) {
    float fbi = (float)bi, fbj = (float)bj;
    v2f b;
    if (hf == 0) { b.x = -2.0f * fbi; b.y = -2.0f * fbj;           }
    else         { b.x = 1.0f;        b.y = fbi * fbi + fbj * fbj; }

    v8f c = {};
    v8f d = __builtin_amdgcn_wmma_f32_16x16x4_f32(
        false, a, false, b, (short)0, c, false, false);

    // relu applied at consume time (one v_max per tile per lane)
    const float wn = __builtin_fmaxf(wlds[sIdx], 0.0f);

    #pragma unroll
    for (int k = 0; k < 8; ++k) {
      float d2    = d[k];
      float sq    = __builtin_amdgcn_sqrtf(d2);
      float tq    = sq + EPSF;                       // sqrt(d2)+eps
      float denom = __builtin_fmaf(tq, tq, z2[k]);   // (sqrt+eps)^2 + z^2
      acc[k] += wn * __builtin_amdgcn_rcpf(denom);
    }

    sIdx += 16;
    bj += 16;
    if (bj >= RES) { bj -= RES; bi += 1; }
  }

  // ---- reduce each row across the 16 source columns (within each half) ----
  #pragma unroll
  for (int k = 0; k < 8; ++k) {
    float v = acc[k];
    v += __shfl_xor(v, 1, 32);
    v += __shfl_xor(v, 2, 32);
    v += __shfl_xor(v, 4, 32);
    v += __shfl_xor(v, 8, 32);
    acc[k] = v;
  }

  // ---- each wave deposits its 16 partial row sums into LDS ----
  if (n < 8) {
    float val = 0.0f;
    #pragma unroll
    for (int k = 0; k < 8; ++k)
      if (n == k) val = acc[k];
    redbuf[wave * 16 + n + 8 * hf] = val;
  }
  __syncthreads();

  // ---- combine the 8 source-splits and store (rim passes boundary through) ----
  if (threadIdx.x < 16) {
    const int m = threadIdx.x;
    float val = 0.0f;
    #pragma unroll
    for (int w2 = 0; w2 < SPLITS; ++w2) val += redbuf[w2 * 16 + m];
    int t = tileT * 16 + m;
    int i = t / RES, j = t - i * RES;
    float di = (float)(i - CENTER), dj = (float)(j - CENTER);
    float r  = __builtin_amdgcn_sqrtf(di * di + dj * dj);
    out[t] = (r >= (float)(CENTER - 1)) ? boundary[t] : val;
  }
}

extern "C" void kernel_launch(void* const* d_in, const int* in_sizes, int n_in,
                              void* d_out, int out_size, void* d_ws, size_t ws_size,
                              hipStream_t stream) {
  const float* boundary = (const float*)d_in[0];
  float* out = (float*)d_out;
  // One block per 16-target tile: 576 blocks x 256 threads (8 wave32 each)
  // = 4608 waves, enough to cover every SIMD32 several times over.
  hipLaunchKernelGGL(holo_wmma_kernel, dim3(TILES), dim3(256), 0, stream,
                     boundary, out);
}